// Model_19799799234919
// MI455X (gfx1250) — compile-verified
//
#include <hip/hip_runtime.h>

typedef __attribute__((ext_vector_type(2))) float v2f;
typedef __attribute__((ext_vector_type(4))) float v4f;
typedef __attribute__((ext_vector_type(8))) float v8f;

#define DIM        64
#define HW         256          // hidden width
#define FS_STRIDE  132          // 128 + 4 pad -> conflict-free LDS banks

// 512 threads = 16 wave32; wave w owns hidden dims [16w, 16w+16).
// Weights live in registers as WMMA A-fragments; grid-stride over 16-edge
// tiles with a TWO-stage register pipeline: features(t) staged to LDS while
// features(t+1) and indices(t+2) are in flight over the WMMA block.
__global__ __launch_bounds__(512) void swiglu_edge_wmma(
    const float* __restrict__ x,     // [65536, 64]
    const float* __restrict__ wg,    // [128, 256] row-major (k, h)
    const float* __restrict__ wi,    // [128, 256]
    const float* __restrict__ wo,    // [256]
    const int*   __restrict__ eidx,  // [2, E]
    float* __restrict__ out,         // [E]
    int E)
{
    __shared__ float Fs[16 * FS_STRIDE];  // feature tile F[edge][k], 8.4 KB
    __shared__ float outAcc[16];

    const int tid  = threadIdx.x;
    const int wave = tid >> 5;            // 0..15 -> hidden-dim slice
    const int lane = tid & 31;
    const int m    = lane & 15;           // M index within 16x16 tile
    const int koff = (lane >> 4) << 1;    // lanes 0-15: k+{0,1}; 16-31: k+{2,3}
    const int hbase = wave << 4;

    // ---- persistent weight A-fragments (loaded once per block) ----
    // A layout for V_WMMA_F32_16X16X4_F32: lane m holds row M=m;
    // VGPR0 = A[m, k0+koff], VGPR1 = A[m, k0+koff+1].
    v2f ag[32], ai[32];
    const int col = hbase + m;            // hidden column of W
#pragma unroll
    for (int s = 0; s < 32; ++s) {
        const int k = (s << 2) + koff;
        ag[s].x = wg[(k    ) * HW + col];
        ag[s].y = wg[(k + 1) * HW + col];
        ai[s].x = wi[(k    ) * HW + col];
        ai[s].y = wi[(k + 1) * HW + col];
    }
    // C/D tile layout: lane n, VGPR r -> M = r + (lane<16 ? 0 : 8), N = n%16.
    float wor[8];
    const int hrow = hbase + ((lane >> 4) << 3);
#pragma unroll
    for (int r = 0; r < 8; ++r) wor[r] = wo[hrow + r];

    const float* fp = &Fs[m * FS_STRIDE + koff];  // per-lane B-fragment base

    // gather geometry (constant per thread): 32 threads/edge, 4 floats each
    const int  e_local = tid >> 5;        // 0..15
    const int  c       = lane << 2;       // 0..124 step 4
    const bool lo_half = (c < DIM);       // scope vs goal half of concat
    const int  scol    = lo_half ? c : (c - DIM);
    const int  ntiles  = E >> 4;          // 16 edges per tile
    const int  stride  = gridDim.x;

    // ---- prologue: features(t0) + resolved index(t1) into registers ----
    v4f vcur;
    int node_next = 0;
    {
        const int e    = (blockIdx.x << 4) + e_local;
        const int node = lo_half ? eidx[e] : eidx[E + e];
        vcur = *(const v4f*)&x[node * DIM + scol];
        const int t1 = blockIdx.x + stride;
        if (t1 < ntiles) {
            const int e1 = (t1 << 4) + e_local;
            node_next = lo_half ? eidx[e1] : eidx[E + e1];
        }
    }

    for (int tile = blockIdx.x; tile < ntiles; tile += stride) {
        if (tid < 16) outAcc[tid] = 0.0f;
        *(v4f*)&Fs[e_local * FS_STRIDE + c] = vcur;   // stage current tile
        __syncthreads();

        // issue BOTH prefetches now; they are consumed only next iteration,
        // so their waits land a full WMMA-block + epilogue after issue.
        const int tn  = tile + stride;                // block-uniform conds
        const int tnn = tn + stride;
        v4f vnext = {};
        int idx_pending = 0;
        if (tn < ntiles)                              // features(t+1):
            vnext = *(const v4f*)&x[node_next * DIM + scol];
        if (tnn < ntiles) {                           // index(t+2):
            const int e = (tnn << 4) + e_local;
            idx_pending = lo_half ? eidx[e] : eidx[E + e];
        }

        // ---- 64 x v_wmma_f32_16x16x4_f32: D[hid16 x edge16] ----
        v8f cg = {};
        v8f cv = {};
#pragma unroll
        for (int s = 0; s < 32; ++s) {
            v2f b;                                    // B = F^T[k0..k0+3, edges]
            b.x = fp[(s << 2)    ];                   // ds_load_2addr_b64 pairs
            b.y = fp[(s << 2) + 1];
            cg = __builtin_amdgcn_wmma_f32_16x16x4_f32(
                    false, ag[s], false, b, (short)0, cg, false, false);
            cv = __builtin_amdgcn_wmma_f32_16x16x4_f32(
                    false, ai[s], false, b, (short)0, cv, false, false);
        }

        // ---- epilogue: sum_h silu(g)*v*wo over this wave's 16 hidden dims --
        float partial = 0.0f;
#pragma unroll
        for (int r = 0; r < 8; ++r) {
            const float g  = cg[r];
            const float sg = g / (1.0f + __expf(-g)); // silu
            partial += sg * cv[r] * wor[r];
        }
        partial += __shfl_xor(partial, 16, 32);       // fold M=0..7 with M=8..15
        if (lane < 16) atomicAdd(&outAcc[lane], partial); // ds_add_f32 x16 waves
        __syncthreads();

        if (tid < 16)                                 // streamed, never re-read
            __builtin_nontemporal_store(outAcc[tid], &out[(tile << 4) + tid]);

        vcur      = vnext;                            // advance pipeline
        node_next = idx_pending;
        __syncthreads();                              // protect Fs/outAcc reuse
    }
}

extern "C" void kernel_launch(void* const* d_in, const int* in_sizes, int n_in,
                              void* d_out, int out_size, void* d_ws, size_t ws_size,
                              hipStream_t stream) {
    const float* x    = (const float*)d_in[0];   // type_reprs [B*N*DIM]
    const float* wg   = (const float*)d_in[1];   // w_gate [128*256]
    const float* wi   = (const float*)d_in[2];   // w_in   [128*256]
    const float* wo   = (const float*)d_in[3];   // w_out  [256]
    const int*   eidx = (const int*)d_in[4];     // edge_index [2*E]
    float* out = (float*)d_out;

    const int E = in_sizes[4] / 2;               // 1,048,576
    const int blocks = 2048;                     // grid-stride over 65536 tiles
    swiglu_edge_wmma<<<blocks, 512, 0, stream>>>(x, wg, wi, wo, eidx, out, E);
}